// MaxTimesPlusOpeningLiftingP4_76390288326909
// MI455X (gfx1250) — compile-verified
//
#include <hip/hip_runtime.h>
#include <stdint.h>

// Problem dims (fixed by reference setup_inputs)
#define B_   16
#define H_   128
#define W_   128
#define C_   3
#define F_   8
#define P_   25
#define CF_  24      // C*F
#define ROT_ 4

// Tiling: 32x16 outputs per block, 256 threads -> 2 output pixels/thread
#define TH 16
#define TW 32
#define EH (TH + 4)          // ero tile rows   (halo +-2)  = 20
#define EW (TW + 4)          // ero tile cols               = 36
#define XH (TH + 8)          // x tile rows     (halo +-4)  = 24
#define XW (TW + 8)          // x tile cols                 = 40
#define ES_STRIDE 25         // padded cf stride (coprime with 64 banks)
#define HP (H_ + 8)          // padded input dims (pad 4 each side)
#define WP (W_ + 8)

// rot90^k source index (jnp.rot90: out[r,s] = in[s, N-1-r])
__device__ __forceinline__ void rotsrc(int k, int r, int s, int& rr, int& ss) {
    switch (k & 3) {
        case 0: rr = r;     ss = s;     break;
        case 1: rr = s;     ss = 4 - r; break;
        case 2: rr = 4 - r; ss = 4 - s; break;
        default:rr = 4 - s; ss = r;     break;
    }
}

// Build rotation-folded constant tables:
//  EA = 1/(tke_rot+eps), EB = -ke_rot*EA   (erosion term = fmaf(y, EA, EB))
//  DA = tk_rot,          DB = k_rot        (dilation term = fmaf(e, DA, DB))
// ke_rot(i) = rot90(kernel[::-1,::-1], i) = rot90(kernel, i+2)
__global__ void prep_tables(const float* __restrict__ kern,
                            const float* __restrict__ tkern,
                            float* __restrict__ EA, float* __restrict__ EB,
                            float* __restrict__ DA, float* __restrict__ DB) {
    int t = blockIdx.x * blockDim.x + threadIdx.x;
    if (t >= ROT_ * P_ * CF_) return;
    int f = t % F_;
    int c = (t / F_) % C_;
    int p = (t / CF_) % P_;
    int i = t / (CF_ * P_);
    int r = p / 5, s = p % 5;
    int rr, ss;

    rotsrc(i, r, s, rr, ss);
    int src = ((rr * 5 + ss) * C_ + c) * F_ + f;
    DA[t] = tkern[src];
    DB[t] = kern[src];

    rotsrc(i + 2, r, s, rr, ss);
    src = ((rr * 5 + ss) * C_ + c) * F_ + f;
    float ea = 1.0f / (tkern[src] + 1e-7f);
    EA[t] = ea;
    EB[t] = -kern[src] * ea;
}

// Zero-padded copy of x: xpad[B][136][136][3], image at offset (+4,+4)
__global__ void pad_x(const float* __restrict__ x, float* __restrict__ xpad) {
    int t = blockIdx.x * blockDim.x + threadIdx.x;
    const int total = B_ * HP * WP * C_;
    if (t >= total) return;
    int c = t % C_;
    int w = (t / C_) % WP;
    int h = (t / (C_ * WP)) % HP;
    int b = t / (C_ * WP * HP);
    int hi = h - 4, wi = w - 4;
    float v = 0.0f;
    if (hi >= 0 && hi < H_ && wi >= 0 && wi < W_)
        v = x[((b * H_ + hi) * W_ + wi) * C_ + c];
    xpad[t] = v;
}

// Fused erosion (with halo) + max-plus dilation per (batch, rotation, tile).
__global__ void __launch_bounds__(256)
fused_open_lift(const float* __restrict__ xpad,
                const float* __restrict__ EA, const float* __restrict__ EB,
                const float* __restrict__ DA, const float* __restrict__ DB,
                float* __restrict__ out) {
    __shared__ __align__(16) float cEA[P_ * CF_], cEB[P_ * CF_];
    __shared__ __align__(16) float cDA[P_ * CF_], cDB[P_ * CF_];
    __shared__ float xs[XH * XW * C_];           // 2880 floats
    __shared__ float es[EH * EW * ES_STRIDE];    // 18000 floats

    const int tid = threadIdx.x;
    const int rot = blockIdx.z & 3;
    const int b   = blockIdx.z >> 2;
    const int h0  = blockIdx.y * TH;
    const int w0  = blockIdx.x * TW;

    // Stage rotation-specific constant tables into LDS (uniform broadcast reads later)
    const int cb = rot * (P_ * CF_);
    for (int t = tid; t < P_ * CF_; t += 256) {
        cEA[t] = EA[cb + t];
        cEB[t] = EB[cb + t];
        cDA[t] = DA[cb + t];
        cDB[t] = DB[cb + t];
    }

    // Async global->LDS copy of the x tile (CDNA5 ASYNCcnt path).
    // Tile covers padded rows [h0, h0+24), cols [w0, w0+40), all channels.
    const int rowlen = XW * C_;  // 120 floats, contiguous in xpad per row
    const uint64_t gpix = ((uint64_t)b * HP + (uint64_t)h0) * WP + (uint64_t)w0;
    for (int t = tid; t < XH * rowlen; t += 256) {
        const int row = t / rowlen;
        const int off = t - row * rowlen;
        const uint64_t gaddr =
            (uint64_t)(uintptr_t)(xpad + (gpix + (uint64_t)row * WP) * C_ + off);
        const uint32_t laddr = (uint32_t)(uintptr_t)(&xs[row * rowlen + off]);
        asm volatile("global_load_async_to_lds_b32 %0, %1, off"
                     :: "v"(laddr), "v"(gaddr) : "memory");
    }
    asm volatile("s_wait_asynccnt 0" ::: "memory");
    __syncthreads();

    // ---- Erosion over the haloed ero tile (EH x EW = 720 pixels, 24 cf each) ----
    for (int e = tid; e < EH * EW; e += 256) {
        const int er = e / EW;
        const int ec = e - er * EW;
        const int ih = h0 + er - 2;    // image coords of this ero pixel
        const int iw = w0 + ec - 2;
        float acc[CF_];
        if (ih >= 0 && ih < H_ && iw >= 0 && iw < W_) {
            #pragma unroll
            for (int j = 0; j < CF_; ++j) acc[j] = 3.402823466e38f;
            #pragma unroll 1
            for (int p = 0; p < P_; ++p) {
                const int ii = p / 5;
                const int jj = p - ii * 5;
                const float* xr = &xs[((er + ii) * XW + (ec + jj)) * C_];
                const float y0 = xr[0], y1 = xr[1], y2 = xr[2];
                const float* A  = &cEA[p * CF_];
                const float* Bn = &cEB[p * CF_];
                #pragma unroll
                for (int f = 0; f < F_; ++f) {
                    acc[f]      = fminf(acc[f],      fmaf(y0, A[f],      Bn[f]));
                    acc[8 + f]  = fminf(acc[8 + f],  fmaf(y1, A[8 + f],  Bn[8 + f]));
                    acc[16 + f] = fminf(acc[16 + f], fmaf(y2, A[16 + f], Bn[16 + f]));
                }
            }
        } else {
            // reference zero-pads ero before the second extract_patches
            #pragma unroll
            for (int j = 0; j < CF_; ++j) acc[j] = 0.0f;
        }
        float* ep = &es[e * ES_STRIDE];
        #pragma unroll
        for (int j = 0; j < CF_; ++j) ep[j] = acc[j];
    }
    __syncthreads();

    // ---- Max-plus dilation: two output pixels per thread (rows orr and orr+8) ----
    // Constant LDS loads per tap are shared between the two pixels.
    const int orr = tid / TW;        // 0..7
    const int occ = tid - orr * TW;  // 0..31
    float acc0[CF_], acc1[CF_];
    #pragma unroll
    for (int j = 0; j < CF_; ++j) { acc0[j] = -3.402823466e38f; acc1[j] = -3.402823466e38f; }
    #pragma unroll 1
    for (int p = 0; p < P_; ++p) {
        const int ii = p / 5;
        const int jj = p - ii * 5;
        const float* ev0 = &es[((orr + ii) * EW + (occ + jj)) * ES_STRIDE];
        const float* ev1 = ev0 + 8 * EW * ES_STRIDE;   // row orr+8+ii
        const float* A   = &cDA[p * CF_];
        const float* Bc  = &cDB[p * CF_];
        #pragma unroll
        for (int j = 0; j < CF_; ++j) {
            const float a = A[j], bb = Bc[j];
            acc0[j] = fmaxf(acc0[j], fmaf(ev0[j], a, bb));
            acc1[j] = fmaxf(acc1[j], fmaf(ev1[j], a, bb));
        }
    }
    // sum over channels; output layout [B, 4, H, W, F]
    float* op0 = out + ((((uint64_t)b * ROT_ + rot) * H_ + (h0 + orr)) * W_ + (w0 + occ)) * F_;
    float* op1 = op0 + (uint64_t)8 * W_ * F_;   // row h0+orr+8
    #pragma unroll
    for (int f = 0; f < F_; ++f) {
        op0[f] = acc0[f] + acc0[8 + f] + acc0[16 + f];
        op1[f] = acc1[f] + acc1[8 + f] + acc1[16 + f];
    }
}

extern "C" void kernel_launch(void* const* d_in, const int* in_sizes, int n_in,
                              void* d_out, int out_size, void* d_ws, size_t ws_size,
                              hipStream_t stream) {
    const float* x  = (const float*)d_in[0];   // [16,128,128,3]
    const float* kw = (const float*)d_in[1];   // [5,5,3,8]
    const float* tk = (const float*)d_in[2];   // [5,5,3,8]
    float* out = (float*)d_out;                // [16,4,128,128,8]

    // Workspace layout: xpad (3.55 MB) + 4 constant tables (38 KB)
    float* xpad = (float*)d_ws;
    float* EA = xpad + (size_t)B_ * HP * WP * C_;
    float* EB = EA + ROT_ * P_ * CF_;
    float* DA = EB + ROT_ * P_ * CF_;
    float* DB = DA + ROT_ * P_ * CF_;

    prep_tables<<<(ROT_ * P_ * CF_ + 255) / 256, 256, 0, stream>>>(kw, tk, EA, EB, DA, DB);

    const int npad = B_ * HP * WP * C_;
    pad_x<<<(npad + 255) / 256, 256, 0, stream>>>(x, xpad);

    dim3 grid(W_ / TW, H_ / TH, B_ * ROT_);   // (4, 8, 64)
    fused_open_lift<<<grid, 256, 0, stream>>>(xpad, EA, EB, DA, DB, out);
}